// BIOULDecoder_29265907155093
// MI455X (gfx1250) — compile-verified
//
#include <hip/hip_runtime.h>

// CRF (BIOUL-constrained) NLL:  mean_b( score_b - logZ_b )
// B=4096 rows, T=2048 steps, 5 tags.
// Time-split forward algorithm:
//   kind 0   : forward vector over t in [0,1024)
//   kind 1..5: basis row i of the 5x5 log-semiring operator of t in [1024,2048)
// Phase B combines halves per row; all emissions/tags/mask staged into LDS via
// gfx1250 async global->LDS b128 copies, double buffered, wave-0 copy crew.

#define IMP       (-10000.0f)
#define T_LEN     2048
#define B_ROWS    4096
#define HALF      1024
#define CHUNK     32
#define NC2       (HALF / CHUNK)      // 32 chunks per half
#define ESTRIDE   164                 // floats per LDS emission stream (160 data + 4 pad)
#define ROWS_PB   16                  // rows per block (x6 kinds = 96 threads)

typedef int   v4i __attribute__((ext_vector_type(4)));
typedef float v2f __attribute__((ext_vector_type(2)));
typedef __attribute__((address_space(1))) v4i gv4i;   // global int4
typedef __attribute__((address_space(3))) v4i lv4i;   // LDS int4

__device__ __forceinline__ void cp16_async(void* lds, const void* g) {
#if defined(__AMDGCN__)
#if __has_builtin(__builtin_amdgcn_global_load_async_to_lds_b128)
  v4i* gp_gen = (v4i*)g;
  v4i* lp_gen = (v4i*)lds;
  __builtin_amdgcn_global_load_async_to_lds_b128((gv4i*)gp_gen, (lv4i*)lp_gen,
                                                 0, 0);
#else
  unsigned l32 = (unsigned)(unsigned long long)(__attribute__((address_space(3))) void*)lds;
  asm volatile("global_load_async_to_lds_b128 %0, %1, off"
               :: "v"(l32), "v"(g) : "memory");
#endif
#endif
}

#if defined(__AMDGCN__)
#if __has_builtin(__builtin_amdgcn_s_wait_asynccnt)
#define WAIT_ASYNC(N) __builtin_amdgcn_s_wait_asynccnt(N)
#else
#define WAIT_ASYNC(N) asm volatile("s_wait_asynccnt %0" :: "i"(N) : "memory")
#endif
#else
#define WAIT_ASYNC(N)
#endif

// Copy one stream's CHUNK slice (640B emis + 128B tags + 32B mask), 50 x b128.
__device__ __forceinline__ void issue_stream(
    const float* emis, const int* tags, const unsigned char* mask,
    float* sE, int* sT, unsigned char* sM, size_t gofs)
{
  const char* gE = (const char*)(emis + gofs * 5);
  char* lE = (char*)sE;
#pragma unroll
  for (int k = 0; k < (CHUNK * 5 * 4) / 16; ++k)      // 40 x b128
    cp16_async(lE + k * 16, gE + k * 16);

  const char* gT = (const char*)(tags + gofs);
  char* lT = (char*)sT;
#pragma unroll
  for (int k = 0; k < (CHUNK * 4) / 16; ++k)          // 8 x b128
    cp16_async(lT + k * 16, gT + k * 16);

  const char* gM = (const char*)(mask + gofs);
  char* lM = (char*)sM;
#pragma unroll
  for (int k = 0; k < CHUNK / 16; ++k)                // 2 x b128
    cp16_async(lM + k * 16, gM + k * 16);
}

__global__ __launch_bounds__(96) void crf_phaseA(
    const float* __restrict__ emis, const unsigned char* __restrict__ mask,
    const int* __restrict__ tags, const float* __restrict__ startT,
    const float* __restrict__ transT, const int* __restrict__ ucp,
    float* __restrict__ wsLp, float* __restrict__ wsSc,
    int* __restrict__ wsLast, int* __restrict__ wsAny)
{
  __shared__ __align__(16) float         sE[2][32 * ESTRIDE];
  __shared__ __align__(16) int           sT[2][32 * CHUNK];
  __shared__ __align__(16) unsigned char sM[2][32 * CHUNK];
  __shared__ float sTr[32];   // constrained raw transitions (25 used)

  const int tid  = threadIdx.x;
  const int rsub = tid / 6;               // 0..15
  const int kind = tid - rsub * 6;        // 0..5
  const int row  = blockIdx.x * ROWS_PB + rsub;
  const int half = (kind > 0) ? 1 : 0;
  const int cs   = rsub * 2 + half;       // consumer stream index
  const int uc   = ucp[0];
  // BIOUL disallowed (i,j) pairs as a 25-bit mask, bit = i*5+j
  const unsigned TM = (1u<<2)|(1u<<4)|(1u<<5)|(1u<<6)|(1u<<8)|(1u<<10)|
                      (1u<<11)|(1u<<13)|(1u<<17)|(1u<<19)|(1u<<22)|(1u<<24);

  float st[5];
  v2f   E01[5], E23[5];
  float E4[5];
#pragma unroll
  for (int j = 0; j < 5; ++j)
    st[j] = (uc && (j == 2 || j == 4)) ? IMP : startT[j];
#pragma unroll
  for (int i = 0; i < 5; ++i) {
    float e_[5];
#pragma unroll
    for (int j = 0; j < 5; ++j) {
      float tv = (uc && ((TM >> (i * 5 + j)) & 1)) ? IMP : transT[i * 5 + j];
      e_[j] = __expf(tv);               // IMPOSSIBLE -> exactly 0
    }
    E01[i].x = e_[0]; E01[i].y = e_[1];
    E23[i].x = e_[2]; E23[i].y = e_[3];
    E4[i] = e_[4];
  }
  if (tid < 25)
    sTr[tid] = (uc && ((TM >> tid) & 1)) ? IMP : transT[tid];

  // state init
  v2f v01 = {0.f, 0.f}, v23 = {0.f, 0.f};
  float v4c = 0.f, c = 0.f, score = 0.f;
  int prev = 0, last = 0, any = 0;
  if (kind > 0) {
    const int b = kind - 1;
    v01.x = (b == 0) ? 1.f : 0.f;  v01.y = (b == 1) ? 1.f : 0.f;
    v23.x = (b == 2) ? 1.f : 0.f;  v23.y = (b == 3) ? 1.f : 0.f;
    v4c   = (b == 4) ? 1.f : 0.f;
    prev = tags[(size_t)row * T_LEN + HALF - 1];   // boundary tag
  }

  // copy crew = wave 0; crew lane t copies stream t (row t>>1, half t&1)
  const bool   crew  = (tid < 32);
  const size_t gBase = ((size_t)(blockIdx.x * ROWS_PB + (tid >> 1))) * T_LEN +
                       (size_t)(tid & 1) * HALF;

  if (crew)
    issue_stream(emis, tags, mask, &sE[0][tid * ESTRIDE], &sT[0][tid * CHUNK],
                 &sM[0][tid * CHUNK], gBase);

  for (int ck = 0; ck < NC2; ++ck) {
    const int buf = ck & 1;
    if (crew) {
      if (ck + 1 < NC2) {
        issue_stream(emis, tags, mask, &sE[buf ^ 1][tid * ESTRIDE],
                     &sT[buf ^ 1][tid * CHUNK], &sM[buf ^ 1][tid * CHUNK],
                     gBase + (size_t)(ck + 1) * CHUNK);
        WAIT_ASYNC(50);   // 50 in flight (next chunk); previous chunk complete
      } else {
        WAIT_ASYNC(0);
      }
#if defined(__AMDGCN__)
      asm volatile("" ::: "memory");
#endif
    }
    __syncthreads();      // chunk ck resident for all 3 waves

    const float*         eRow = &sE[buf][cs * ESTRIDE];
    const int*           tRow = &sT[buf][cs * CHUNK];
    const unsigned char* mRow = &sM[buf][cs * CHUNK];

#pragma unroll 4
    for (int s = 0; s < CHUNK; ++s) {
      const int tGlob = half * HALF + ck * CHUNK + s;
      float e0 = eRow[s * 5 + 0], e1 = eRow[s * 5 + 1], e2 = eRow[s * 5 + 2],
            e3 = eRow[s * 5 + 3], e4 = eRow[s * 5 + 4];
      const int tg = tRow[s];
      const float esel = eRow[s * 5 + tg];        // LDS gather, replaces cndmask chain
      float F0 = __expf(e0), F1 = __expf(e1), F2 = __expf(e2),
            F3 = __expf(e3), F4 = __expf(e4);

      if (tGlob == 0) {               // only kind 0, first step
        v01.x = __expf(st[0] + e0); v01.y = __expf(st[1] + e1);
        v23.x = __expf(st[2] + e2); v23.y = __expf(st[3] + e3);
        v4c   = __expf(st[4] + e4);
        float ssel = st[0];
        ssel = (tg == 1) ? st[1] : ssel;
        ssel = (tg == 2) ? st[2] : ssel;
        ssel = (tg == 3) ? st[3] : ssel;
        ssel = (tg == 4) ? st[4] : ssel;
        score = ssel + esel;
        prev = tg; last = tg; any = 1; c = 0.f;
      } else {
        const bool mt = (mRow[s] != 0);
        const float a0 = v01.x, a1 = v01.y, a2 = v23.x, a3 = v23.y, a4 = v4c;
        v2f w;
        // n01/n23 via packed f32 FMA; n4 scalar
        w.x = a0; w.y = a0;
        v2f n01 = w * E01[0];
        v2f n23 = w * E23[0];
        float n4 = a0 * E4[0];
        w.x = a1; w.y = a1;
        n01 = __builtin_elementwise_fma(w, E01[1], n01);
        n23 = __builtin_elementwise_fma(w, E23[1], n23);
        n4  = fmaf(a1, E4[1], n4);
        w.x = a2; w.y = a2;
        n01 = __builtin_elementwise_fma(w, E01[2], n01);
        n23 = __builtin_elementwise_fma(w, E23[2], n23);
        n4  = fmaf(a2, E4[2], n4);
        w.x = a3; w.y = a3;
        n01 = __builtin_elementwise_fma(w, E01[3], n01);
        n23 = __builtin_elementwise_fma(w, E23[3], n23);
        n4  = fmaf(a3, E4[3], n4);
        w.x = a4; w.y = a4;
        n01 = __builtin_elementwise_fma(w, E01[4], n01);
        n23 = __builtin_elementwise_fma(w, E23[4], n23);
        n4  = fmaf(a4, E4[4], n4);
        v2f f01; f01.x = F0; f01.y = F1;
        v2f f23; f23.x = F2; f23.y = F3;
        n01 *= f01; n23 *= f23; n4 *= F4;
        v01 = mt ? n01 : v01;
        v23 = mt ? n23 : v23;
        v4c = mt ? n4  : v4c;
        const float tr = sTr[prev * 5 + tg];
        score += mt ? (tr + esel) : 0.f;
        prev = tg;
        last = mt ? tg : last;
        any  = mt ? 1 : any;
      }
      if ((s & 3) == 3) {             // renormalize every 4 steps
        float mx = fmaxf(fmaxf(fmaxf(v01.x, v01.y), fmaxf(v23.x, v23.y)), v4c);
        float r  = __builtin_amdgcn_rcpf(mx);
        v01.x *= r; v01.y *= r; v23.x *= r; v23.y *= r; v4c *= r;
        c += __logf(mx);
      }
    }
    __syncthreads();      // everyone done with buf before crew overwrites it
  }

  // emit half-results: lp[5] (log of scaled vector), score, last, any
  const size_t o = (size_t)row * 6 + kind;
  float vv[5] = {v01.x, v01.y, v23.x, v23.y, v4c};
#pragma unroll
  for (int j = 0; j < 5; ++j)
    wsLp[o * 5 + j] = (vv[j] <= 0.f) ? -3.0e38f : (c + __logf(vv[j]));
  wsSc[o]   = score;
  wsLast[o] = last;
  wsAny[o]  = any;
}

__global__ __launch_bounds__(32) void crf_phaseB(
    const float* __restrict__ endT, const int* __restrict__ ucp,
    const float* __restrict__ wsLp, const float* __restrict__ wsSc,
    const int* __restrict__ wsLast, const int* __restrict__ wsAny,
    float* __restrict__ partial)
{
  const int row = blockIdx.x * 32 + threadIdx.x;
  const int uc  = ucp[0];
  float en[5];
#pragma unroll
  for (int j = 0; j < 5; ++j)
    en[j] = (uc && (j == 1 || j == 2)) ? IMP : endT[j];

  float lp0[5];
#pragma unroll
  for (int i = 0; i < 5; ++i) lp0[i] = wsLp[((size_t)row * 6 + 0) * 5 + i];

  // lp_final[j] = LSE_i( lp0[i] + A[i][j] ),  A row i = kind (i+1)
  float lpf[5];
#pragma unroll
  for (int j = 0; j < 5; ++j) {
    float tv[5];
    float m = -3.0e38f;
#pragma unroll
    for (int i = 0; i < 5; ++i) {
      tv[i] = lp0[i] + wsLp[((size_t)row * 6 + 1 + i) * 5 + j];
      m = fmaxf(m, tv[i]);
    }
    if (m > -1.0e30f) {
      float ssum = 0.f;
#pragma unroll
      for (int i = 0; i < 5; ++i) ssum += __expf(tv[i] - m);
      lpf[j] = m + __logf(ssum);
    } else {
      lpf[j] = -3.0e38f;
    }
  }
  // z = LSE_j( lp_final[j] + end[j] )
  float t[5], m2 = -3.0e38f;
#pragma unroll
  for (int j = 0; j < 5; ++j) { t[j] = lpf[j] + en[j]; m2 = fmaxf(m2, t[j]); }
  float s2 = 0.f;
#pragma unroll
  for (int j = 0; j < 5; ++j) s2 += __expf(t[j] - m2);
  const float z = m2 + __logf(s2);

  const float score = wsSc[(size_t)row * 6 + 0] + wsSc[(size_t)row * 6 + 1];
  const int last = wsAny[(size_t)row * 6 + 1] ? wsLast[(size_t)row * 6 + 1]
                                              : wsLast[(size_t)row * 6 + 0];
  float endsel = en[0];
  endsel = (last == 1) ? en[1] : endsel;
  endsel = (last == 2) ? en[2] : endsel;
  endsel = (last == 3) ? en[3] : endsel;
  endsel = (last == 4) ? en[4] : endsel;

  float nll = (score + endsel) - z;
#pragma unroll
  for (int m = 16; m >= 1; m >>= 1)
    nll += __shfl_xor(nll, m, 32);
  if (threadIdx.x == 0) partial[blockIdx.x] = nll;
}

__global__ __launch_bounds__(128) void reduce128(const float* __restrict__ part,
                                                 float* __restrict__ out)
{
  __shared__ float sh[128];
  const int t = threadIdx.x;
  sh[t] = part[t];
  __syncthreads();
#pragma unroll
  for (int w = 64; w >= 1; w >>= 1) {
    if (t < w) sh[t] += sh[t + w];
    __syncthreads();
  }
  if (t == 0) out[0] = sh[0] * (1.0f / (float)B_ROWS);
}

extern "C" void kernel_launch(void* const* d_in, const int* in_sizes, int n_in,
                              void* d_out, int out_size, void* d_ws, size_t ws_size,
                              hipStream_t stream)
{
  (void)in_sizes; (void)n_in; (void)out_size; (void)ws_size;
  const float*         emis   = (const float*)d_in[0];
  const unsigned char* mask   = (const unsigned char*)d_in[1];
  const int*           tags   = (const int*)d_in[2];
  const float*         startT = (const float*)d_in[3];
  const float*         transT = (const float*)d_in[4];
  const float*         endT   = (const float*)d_in[5];
  const int*           uc     = (const int*)d_in[6];

  float* wsLp    = (float*)d_ws;                 // 4096*6*5
  float* wsSc    = wsLp + (size_t)B_ROWS * 6 * 5;// 4096*6
  int*   wsLast  = (int*)(wsSc + (size_t)B_ROWS * 6);
  int*   wsAny   = wsLast + (size_t)B_ROWS * 6;
  float* partial = (float*)(wsAny + (size_t)B_ROWS * 6);  // 128

  crf_phaseA<<<B_ROWS / ROWS_PB, 96, 0, stream>>>(emis, mask, tags, startT,
                                                  transT, uc, wsLp, wsSc,
                                                  wsLast, wsAny);
  crf_phaseB<<<B_ROWS / 32, 32, 0, stream>>>(endT, uc, wsLp, wsSc, wsLast,
                                             wsAny, partial);
  reduce128<<<1, 128, 0, stream>>>(partial, (float*)d_out);
}